// BiGATFusionModel_42434276885024
// MI455X (gfx1250) — compile-verified
//
#include <hip/hip_runtime.h>

// MI455X / gfx1250, wave32. fp32 WMMA GEMM + GAT edge-softmax aggregation.

typedef float v2f __attribute__((ext_vector_type(2)));
typedef float v8f __attribute__((ext_vector_type(8)));

static constexpr int DIM = 128;

// ---------------------------------------------------------------------------
// Zero-fill workspace region (accumulators + softmax denominators).
// ---------------------------------------------------------------------------
__global__ void zero_kernel(float* __restrict__ p, long n) {
  long i = (long)blockIdx.x * blockDim.x + threadIdx.x;
  const long stride = (long)gridDim.x * blockDim.x;
  for (; i < n; i += stride) p[i] = 0.0f;
}

// ---------------------------------------------------------------------------
// H = X @ W  (X: nrows x 128 row-major, W: 128 x 128 row-major) using
// V_WMMA_F32_16X16X4_F32. One wave computes a 16-row strip x 128 cols
// (8 accumulator tiles), K stepped by 4.
//
// W is staged in LDS in a pair-interleaved layout:
//   sWp[(k>>1)*PSTRIDE + 2*n + (k&1)] = W[k][n]
// so each lane's B-fragment (W[kb][n], W[kb+1][n]) is one aligned
// ds_load_b64 straight into an even VGPR pair (no regrouping movs).
// PSTRIDE = 288 (=256+32 dwords) makes the two half-waves (K-half 0/1,
// +1 pair-block apart) hit disjoint 32-bank sets of the 64 LDS banks.
//
// A-frag (16x4 f32): lane L holds M = L%16; VGPR j holds K = 2*(L/16)+j
//   -> one float2 global load per K-step.
// C/D (16x16 f32, 8 VGPRs): lane L holds N = L%16; VGPR r holds
//   M = r + 8*(L/16).
// ---------------------------------------------------------------------------
__global__ void __launch_bounds__(256) gemm128_wmma_f32(
    const float* __restrict__ X, const float* __restrict__ W,
    float* __restrict__ H, int nrows) {
  constexpr int PSTRIDE = 288;         // dwords per K-pair block (padded)
  __shared__ float sWp[64 * PSTRIDE];  // 72 KB of the 320 KB WGP LDS

  const int tid = threadIdx.x;
  {
    const float4* Wv = (const float4*)W;
#pragma unroll
    for (int i = 0; i < 16; ++i) {     // 256 thr x 16 = 4096 float4 = 128x128
      const int idx = tid + i * 256;
      const int k = idx >> 5;          // source row 0..127
      const int n = (idx & 31) * 4;    // source col
      const float4 w = Wv[idx];
      float* d = sWp + (k >> 1) * PSTRIDE + (k & 1);
      d[2 * (n + 0)] = w.x;
      d[2 * (n + 1)] = w.y;
      d[2 * (n + 2)] = w.z;
      d[2 * (n + 3)] = w.w;
    }
  }
  __syncthreads();

  const int lane = tid & 31;
  const int wave = tid >> 5;
  const int m0 = (blockIdx.x * 8 + wave) * 16;
  if (m0 >= nrows) return;  // wave-uniform: EXEC stays all-ones for WMMA

  const int ml = lane & 15;  // M (for A) / N (for B,C,D) within tile
  const int kh = lane >> 4;  // K-half selector
  int mr = m0 + ml;
  if (mr >= nrows) mr = nrows - 1;  // per-lane clamp (select, no branch)
  const float* xrow = X + (size_t)mr * DIM;

  v8f acc[8];
#pragma unroll
  for (int t = 0; t < 8; ++t)
#pragma unroll
    for (int r = 0; r < 8; ++r) acc[t][r] = 0.0f;

#pragma unroll 2
  for (int k = 0; k < DIM; k += 4) {
    const int kb = k + 2 * kh;
    const float2 a2 = *(const float2*)(xrow + kb);
    v2f av;
    av[0] = a2.x;
    av[1] = a2.y;
    const float* brow = sWp + (kb >> 1) * PSTRIDE;
#pragma unroll
    for (int t = 0; t < 8; ++t) {
      const float2 b2 = *(const float2*)(brow + 2 * (t * 16 + ml));
      v2f bv;
      bv[0] = b2.x;
      bv[1] = b2.y;
      acc[t] = __builtin_amdgcn_wmma_f32_16x16x4_f32(
          false, av, false, bv, (short)0, acc[t], false, false);
    }
  }

#pragma unroll
  for (int t = 0; t < 8; ++t) {
#pragma unroll
    for (int r = 0; r < 8; ++r) {
      const int row = m0 + r + 8 * kh;
      if (row < nrows) H[(size_t)row * DIM + t * 16 + ml] = acc[t][r];
    }
  }
}

// ---------------------------------------------------------------------------
// s[node] = dot(H[node, :], a[0:128]) — one wave per node, xor-reduce.
// ---------------------------------------------------------------------------
__global__ void __launch_bounds__(256) proj_kernel(
    const float* __restrict__ H, const float* __restrict__ a,
    float* __restrict__ s, int n) {
  const int lane = threadIdx.x & 31;
  const int node = blockIdx.x * (blockDim.x >> 5) + (threadIdx.x >> 5);
  if (node >= n) return;
  const float4 h = ((const float4*)(H + (size_t)node * DIM))[lane];
  const float4 av = ((const float4*)a)[lane];
  float v = h.x * av.x + h.y * av.y + h.z * av.z + h.w * av.w;
#pragma unroll
  for (int off = 16; off > 0; off >>= 1) v += __shfl_xor(v, off, 32);
  if (lane == 0) s[node] = v;
}

// ---------------------------------------------------------------------------
// Pass 1: ex[i] = exp(leaky_relu(s_dst[dst] + s_src[src])); denom[dst] += ex.
// Global-max subtraction dropped (softmax shift invariance; |e| is small).
// ---------------------------------------------------------------------------
__global__ void __launch_bounds__(256) edge_exp_kernel(
    const float* __restrict__ s_dst, const float* __restrict__ s_src,
    const int* __restrict__ src, const int* __restrict__ dst,
    float* __restrict__ ex, float* __restrict__ denom, int E) {
  const int i = blockIdx.x * blockDim.x + threadIdx.x;
  if (i >= E) return;
  const int d = dst[i];
  float e = s_dst[d] + s_src[src[i]];
  e = (e > 0.0f) ? e : 0.2f * e;  // leaky_relu, slope 0.2
  const float x = __expf(e);
  ex[i] = x;
  atomicAdd(&denom[d], x);
}

// ---------------------------------------------------------------------------
// Pass 2: out[dst] += (ex/denom[dst]) * Hsrc[src].  One wave per edge:
// float4 gather per lane + 4 global f32 atomic adds.
// ---------------------------------------------------------------------------
__global__ void __launch_bounds__(256) edge_agg_kernel(
    const float* __restrict__ ex, const float* __restrict__ denom,
    const float* __restrict__ Hsrc, const int* __restrict__ src,
    const int* __restrict__ dst, float* __restrict__ out, int E) {
  const int lane = threadIdx.x & 31;
  const int e = blockIdx.x * (blockDim.x >> 5) + (threadIdx.x >> 5);
  if (e >= E) return;
  const int s = src[e];
  const int d = dst[e];
  const float alpha = ex[e] / (denom[d] + 1e-16f);
  const float4 h = ((const float4*)(Hsrc + (size_t)s * DIM))[lane];
  float* o = out + (size_t)d * DIM + lane * 4;
  atomicAdd(o + 0, alpha * h.x);
  atomicAdd(o + 1, alpha * h.y);
  atomicAdd(o + 2, alpha * h.z);
  atomicAdd(o + 3, alpha * h.w);
}

// ---------------------------------------------------------------------------
// Fusion: f = relu(feat), t = relu(topo);
// g = sigmoid(dot(f, gW[0:128]) + dot(t, gW[128:256]) + b);
// out = g*f + (1-g)*t.  One wave per node.
// ---------------------------------------------------------------------------
__global__ void __launch_bounds__(256) fuse_kernel(
    const float* __restrict__ feat, const float* __restrict__ topo,
    const float* __restrict__ gW, const float* __restrict__ gb,
    float* __restrict__ out, int n) {
  const int lane = threadIdx.x & 31;
  const int node = blockIdx.x * (blockDim.x >> 5) + (threadIdx.x >> 5);
  if (node >= n) return;
  float4 f = ((const float4*)(feat + (size_t)node * DIM))[lane];
  float4 t = ((const float4*)(topo + (size_t)node * DIM))[lane];
  f.x = fmaxf(f.x, 0.0f); f.y = fmaxf(f.y, 0.0f);
  f.z = fmaxf(f.z, 0.0f); f.w = fmaxf(f.w, 0.0f);
  t.x = fmaxf(t.x, 0.0f); t.y = fmaxf(t.y, 0.0f);
  t.z = fmaxf(t.z, 0.0f); t.w = fmaxf(t.w, 0.0f);
  const float4 w0 = ((const float4*)gW)[lane];
  const float4 w1 = ((const float4*)gW)[lane + 32];
  float v = f.x * w0.x + f.y * w0.y + f.z * w0.z + f.w * w0.w +
            t.x * w1.x + t.y * w1.y + t.z * w1.z + t.w * w1.w;
#pragma unroll
  for (int off = 16; off > 0; off >>= 1) v += __shfl_xor(v, off, 32);
  const float g = 1.0f / (1.0f + __expf(-(v + gb[0])));
  float4 o;
  o.x = g * f.x + (1.0f - g) * t.x;
  o.y = g * f.y + (1.0f - g) * t.y;
  o.z = g * f.z + (1.0f - g) * t.z;
  o.w = g * f.w + (1.0f - g) * t.w;
  ((float4*)(out + (size_t)node * DIM))[lane] = o;
}

// ---------------------------------------------------------------------------
extern "C" void kernel_launch(void* const* d_in, const int* in_sizes, int n_in,
                              void* d_out, int out_size, void* d_ws,
                              size_t ws_size, hipStream_t stream) {
  const float* drug_emb = (const float*)d_in[0];
  const float* dis_emb = (const float*)d_in[1];
  const float* Wd_feat = (const float*)d_in[2];
  const float* ad_feat = (const float*)d_in[3];
  const float* Wp_feat = (const float*)d_in[4];
  const float* ap_feat = (const float*)d_in[5];
  const float* W_drug = (const float*)d_in[6];
  const float* W_dis = (const float*)d_in[7];
  const float* a_drug = (const float*)d_in[8];
  const float* a_dis = (const float*)d_in[9];
  const float* gate_drug_W = (const float*)d_in[10];
  const float* gate_drug_b = (const float*)d_in[11];
  const float* gate_dis_W = (const float*)d_in[12];
  const float* gate_dis_b = (const float*)d_in[13];
  const int* df_src = (const int*)d_in[14];
  const int* df_dst = (const int*)d_in[15];
  const int* pf_src = (const int*)d_in[16];
  const int* pf_dst = (const int*)d_in[17];
  const int* d2r_src = (const int*)d_in[18];  // dis_to_drug
  const int* d2r_dst = (const int*)d_in[19];
  const int* r2d_src = (const int*)d_in[20];  // drug_to_dis
  const int* r2d_dst = (const int*)d_in[21];

  const int n_drug = in_sizes[0] / DIM;
  const int n_dis = in_sizes[1] / DIM;
  const int E = in_sizes[14];

  const size_t NBd = (size_t)n_drug * DIM;
  const size_t NBp = (size_t)n_dis * DIM;

  // Workspace layout (~210 MB for the reference sizes):
  float* ws = (float*)d_ws;
  float* H1 = ws;             // h_d_feat  [n_drug x 128]
  float* H2 = H1 + NBd;       // h_p_feat  [n_dis  x 128]
  float* H3 = H2 + NBp;       // h_drug    [n_drug x 128]
  float* H4 = H3 + NBd;       // h_dis     [n_dis  x 128]
  float* F1 = H4 + NBp;       // drug_feat [n_drug x 128]  (atomic acc)
  float* F2 = F1 + NBd;       // dis_feat  [n_dis  x 128]
  float* F3 = F2 + NBp;       // drug_topo [n_drug x 128]
  float* F4 = F3 + NBd;       // dis_topo  [n_dis  x 128]
  float* denA = F4 + NBp;     // denominators: A(drug) B(dis) C(drug) D(dis)
  float* denB = denA + n_drug;
  float* denC = denB + n_dis;
  float* denD = denC + n_drug;
  float* sAd = denD + n_dis;  // per-node attention scalars
  float* sAs = sAd + n_drug;
  float* sBd = sAs + n_drug;
  float* sBs = sBd + n_dis;
  float* sCd = sBs + n_dis;   // graph C: dst over drug, src over dis
  float* sCs = sCd + n_drug;
  float* sDd = sCs + n_dis;   // graph D: dst over dis, src over drug
  float* sDs = sDd + n_dis;
  float* exb = sDs + n_drug;  // E floats, reused per graph (stream-serialized)

  // Zero accumulators + denominators (F1..F4 and den* are contiguous).
  const long zcount = (long)(2 * NBd + 2 * NBp) + 2L * (n_drug + n_dis);
  zero_kernel<<<2048, 256, 0, stream>>>(F1, zcount);

  const dim3 blk(256);
  auto gemm_grid = [](int nrows) {
    const int strips = (nrows + 15) / 16;
    return (strips + 7) / 8;
  };
  gemm128_wmma_f32<<<gemm_grid(n_drug), blk, 0, stream>>>(drug_emb, Wd_feat, H1, n_drug);
  gemm128_wmma_f32<<<gemm_grid(n_dis), blk, 0, stream>>>(dis_emb, Wp_feat, H2, n_dis);
  gemm128_wmma_f32<<<gemm_grid(n_drug), blk, 0, stream>>>(drug_emb, W_drug, H3, n_drug);
  gemm128_wmma_f32<<<gemm_grid(n_dis), blk, 0, stream>>>(dis_emb, W_dis, H4, n_dis);

  auto wgrid = [](int n) { return (n + 7) / 8; };  // 8 waves / 256-thread block
  proj_kernel<<<wgrid(n_drug), blk, 0, stream>>>(H1, ad_feat, sAd, n_drug);
  proj_kernel<<<wgrid(n_drug), blk, 0, stream>>>(H1, ad_feat + DIM, sAs, n_drug);
  proj_kernel<<<wgrid(n_dis), blk, 0, stream>>>(H2, ap_feat, sBd, n_dis);
  proj_kernel<<<wgrid(n_dis), blk, 0, stream>>>(H2, ap_feat + DIM, sBs, n_dis);
  proj_kernel<<<wgrid(n_drug), blk, 0, stream>>>(H3, a_drug, sCd, n_drug);
  proj_kernel<<<wgrid(n_dis), blk, 0, stream>>>(H4, a_drug + DIM, sCs, n_dis);
  proj_kernel<<<wgrid(n_dis), blk, 0, stream>>>(H4, a_dis, sDd, n_dis);
  proj_kernel<<<wgrid(n_drug), blk, 0, stream>>>(H3, a_dis + DIM, sDs, n_drug);

  const int egrid = (E + 255) / 256;
  const int agrid = (E + 7) / 8;

  // A: drug feature graph (src/dst over drugs, features H1)
  edge_exp_kernel<<<egrid, blk, 0, stream>>>(sAd, sAs, df_src, df_dst, exb, denA, E);
  edge_agg_kernel<<<agrid, blk, 0, stream>>>(exb, denA, H1, df_src, df_dst, F1, E);
  // B: disease feature graph
  edge_exp_kernel<<<egrid, blk, 0, stream>>>(sBd, sBs, pf_src, pf_dst, exb, denB, E);
  edge_agg_kernel<<<agrid, blk, 0, stream>>>(exb, denB, H2, pf_src, pf_dst, F2, E);
  // C: dis -> drug (src feats = h_dis = H4)
  edge_exp_kernel<<<egrid, blk, 0, stream>>>(sCd, sCs, d2r_src, d2r_dst, exb, denC, E);
  edge_agg_kernel<<<agrid, blk, 0, stream>>>(exb, denC, H4, d2r_src, d2r_dst, F3, E);
  // D: drug -> dis (src feats = h_drug = H3)
  edge_exp_kernel<<<egrid, blk, 0, stream>>>(sDd, sDs, r2d_src, r2d_dst, exb, denD, E);
  edge_agg_kernel<<<agrid, blk, 0, stream>>>(exb, denD, H3, r2d_src, r2d_dst, F4, E);

  float* out = (float*)d_out;
  fuse_kernel<<<wgrid(n_drug), blk, 0, stream>>>(F1, F3, gate_drug_W, gate_drug_b, out, n_drug);
  fuse_kernel<<<wgrid(n_dis), blk, 0, stream>>>(F2, F4, gate_dis_W, gate_dis_b, out + NBd, n_dis);
}